// MessagePassingEncoder_27642409517239
// MI455X (gfx1250) — compile-verified
//
#include <hip/hip_runtime.h>
#include <cstdint>

#define EMBED_DIM 128

// ---------------------------------------------------------------- utilities
__global__ void zero_f32_kernel(float* __restrict__ p, int n) {
    int i = blockIdx.x * blockDim.x + threadIdx.x;
    if (i < n) p[i] = 0.0f;
}

// deg[col[e]] += 1  (destination-node degree over real edges; self-loop added later)
__global__ void degree_kernel(const int* __restrict__ col, float* __restrict__ deg, int E) {
    int e = blockIdx.x * blockDim.x + threadIdx.x;
    if (e < E) atomicAdd(&deg[col[e]], 1.0f);
}

// deg -> deg_inv_sqrt, folding in the self loop (+1). deg+1 >= 1 so always > 0.
__global__ void dinv_kernel(float* __restrict__ deg, int n) {
    int i = blockIdx.x * blockDim.x + threadIdx.x;
    if (i < n) deg[i] = rsqrtf(deg[i] + 1.0f);
}

// out[i]  = bias + dinv[i]^2 * x[i]   (bias + self-loop message)
// out2[i] = x[i]                      (init_embeds passthrough output)
// one thread = one float4 of one node's row (32 lanes cover 128 dims)
__global__ __launch_bounds__(256) void init_out_kernel(
    const float* __restrict__ x, const float* __restrict__ bias,
    const float* __restrict__ dinv, float* __restrict__ out,
    float* __restrict__ out2, int n) {
    int gid  = blockIdx.x * blockDim.x + threadIdx.x;
    int node = gid >> 5;
    int lane = gid & 31;
    if (node >= n) return;
    float di = dinv[node];
    float w  = di * di;
    size_t off = (size_t)node * EMBED_DIM + lane * 4;
    const float4 xv = *reinterpret_cast<const float4*>(x + off);
    const float4 bv = *reinterpret_cast<const float4*>(bias + lane * 4);
    float4 o;
    o.x = fmaf(w, xv.x, bv.x);
    o.y = fmaf(w, xv.y, bv.y);
    o.z = fmaf(w, xv.z, bv.z);
    o.w = fmaf(w, xv.w, bv.w);
    *reinterpret_cast<float4*>(out + off)  = o;
    *reinterpret_cast<float4*>(out2 + off) = xv;
}

// ---------------------------------------------------------------- edge scatter
// 256 edges per block. Edge indices are staged into LDS with CDNA5 async
// global->LDS loads (ASYNCcnt-tracked), then each of the 8 waves processes
// 32 edges: one wave per edge, lane l handles dims [4l, 4l+3] so the x[row]
// gather is a single contiguous 512B wave read; indices are lifted to SGPRs
// (readfirstlane) so dinv lookups become scalar loads; results go to the
// L2-resident output via no-return global_atomic_add_f32. The inner loop has
// a precomputed trip count so the compiler can unroll and overlap the next
// edge's gather with the previous edge's (return-free) atomics.
__global__ __launch_bounds__(256) void edge_scatter_kernel(
    const int* __restrict__ ei,          // [2*E]: rows then cols
    const float* __restrict__ x,
    const float* __restrict__ dinv,
    float* __restrict__ out, int E) {
    __shared__ int sRow[256];
    __shared__ int sCol[256];

    const int t    = threadIdx.x;
    const int base = blockIdx.x * 256;
    int e  = base + t;
    int ec = (e < E) ? e : (E - 1);   // clamp so every lane loads valid memory

    // LDS byte offsets: generic pointers to LDS carry the offset in the low 32 bits
    unsigned int ldsRow = (unsigned int)(uintptr_t)&sRow[t];
    unsigned int ldsCol = (unsigned int)(uintptr_t)&sCol[t];
    unsigned long long gRow = (unsigned long long)(uintptr_t)(ei + ec);
    unsigned long long gCol = (unsigned long long)(uintptr_t)(ei + E + ec);

    // CDNA5 async memory path: GLOBAL_LOAD_ASYNC_TO_LDS_B32 + S_WAIT_ASYNCCNT
    asm volatile("global_load_async_to_lds_b32 %0, %1, off"
                 :: "v"(ldsRow), "v"(gRow) : "memory");
    asm volatile("global_load_async_to_lds_b32 %0, %1, off"
                 :: "v"(ldsCol), "v"(gCol) : "memory");
    asm volatile("s_wait_asynccnt 0" ::: "memory");
    __syncthreads();

    const int wave  = t >> 5;
    const int lane  = t & 31;
    const int start = wave << 5;

    int nE = E - base - start;        // edges this wave actually owns
    if (nE > 32) nE = 32;             // (may be <= 0 for idle tail waves)

    for (int i = 0; i < nE; ++i) {
        int idx = start + i;
        // wave-uniform indices -> SGPRs -> scalar dinv loads
        int r = __builtin_amdgcn_readfirstlane(sRow[idx]);
        int c = __builtin_amdgcn_readfirstlane(sCol[idx]);
        float norm = dinv[r] * dinv[c];

        const float4 xv =
            *reinterpret_cast<const float4*>(x + (size_t)r * EMBED_DIM + lane * 4);

        // prefetch the next edge's source row while this edge's atomics drain
        int nidx = idx + ((i + 1 < nE) ? 1 : 0);   // clamp: always a valid LDS slot
        int rn   = sRow[nidx];
        __builtin_prefetch(x + (size_t)rn * EMBED_DIM + lane * 4, 0, 0);

        float* op = out + (size_t)c * EMBED_DIM + lane * 4;
        atomicAdd(op + 0, norm * xv.x);
        atomicAdd(op + 1, norm * xv.y);
        atomicAdd(op + 2, norm * xv.z);
        atomicAdd(op + 3, norm * xv.w);
    }
}

// ---------------------------------------------------------------- launcher
extern "C" void kernel_launch(void* const* d_in, const int* in_sizes, int n_in,
                              void* d_out, int out_size, void* d_ws, size_t ws_size,
                              hipStream_t stream) {
    const float* x    = (const float*)d_in[0];
    const int*   ei   = (const int*)d_in[1];   // [2, E] int32, rows then cols
    const float* bias = (const float*)d_in[2];

    const int N = in_sizes[0] / EMBED_DIM;     // 50000
    const int E = in_sizes[1] / 2;             // 600000

    float* out  = (float*)d_out;                         // [N, 128]
    float* out2 = out + (size_t)N * EMBED_DIM;           // init_embeds copy
    float* deg  = (float*)d_ws;                          // N floats of scratch

    dim3 blk(256);
    zero_f32_kernel<<<dim3((N + 255) / 256), blk, 0, stream>>>(deg, N);
    degree_kernel<<<dim3((E + 255) / 256), blk, 0, stream>>>(ei + E, deg, E);
    dinv_kernel<<<dim3((N + 255) / 256), blk, 0, stream>>>(deg, N);
    init_out_kernel<<<dim3((N * 32 + 255) / 256), blk, 0, stream>>>(
        x, bias, deg, out, out2, N);
    edge_scatter_kernel<<<dim3((E + 255) / 256), blk, 0, stream>>>(
        ei, x, deg, out, E);
}